// SCPositionwiseFeedForward_low_82317343195169
// MI455X (gfx1250) — compile-verified
//
#include <hip/hip_runtime.h>
#include <math.h>

typedef __attribute__((ext_vector_type(2))) float v2f;
typedef __attribute__((ext_vector_type(8))) float v8f;

#define B_  8
#define T_  1024
#define DIN 256
#define DH1 1024
#define DH2 1024
#define KW  9

__device__ __forceinline__ float mishf(float v) {
    float sp = (v > 20.f) ? v : log1pf(__expf(v));
    return v * tanhf(sp);
}

// ---------------- norm kernels ----------------
__global__ void k_norm_d(const float* __restrict__ dw, float* __restrict__ inv_nd) {
    int tid = threadIdx.x;
    if (tid >= B_ * KW) return;
    int b = tid / KW, k = tid % KW;
    float acc = 0.f;
    for (int c = 0; c < DH1; ++c) {
        float v = dw[((size_t)b * DH1 + c) * KW + k];
        acc += v * v;
    }
    inv_nd[tid] = 1.f / fmaxf(sqrtf(acc), 1e-12f);
}

__global__ __launch_bounds__(256) void k_norm_p(const float* __restrict__ pw,
                                                float* __restrict__ inv_np) {
    __shared__ float s[256];
    int b  = blockIdx.x >> 4;
    int o  = ((blockIdx.x & 15) << 6) + (threadIdx.x & 63);
    int cg = threadIdx.x >> 6;
    const float* P = pw + (size_t)b * DH1 * DH2;
    float acc = 0.f;
    for (int c = cg; c < DH1; c += 4) {
        float v = P[(size_t)c * DH2 + o];
        acc += v * v;
    }
    s[threadIdx.x] = acc;
    __syncthreads();
    if (cg == 0) {
        float tot = s[threadIdx.x] + s[threadIdx.x + 64] + s[threadIdx.x + 128] + s[threadIdx.x + 192];
        inv_np[b * DH2 + o] = 1.f / fmaxf(sqrtf(tot), 1e-12f);
    }
}

// ---------------- GEMM 1: h = mish(w1 @ x^T + b1), h[b][c][t] ----------------
__global__ __launch_bounds__(256) void k_gemm1_mish(
    const float* __restrict__ x, const float* __restrict__ w1w,
    const float* __restrict__ w1b, float* __restrict__ h)
{
    const int lane = threadIdx.x & 31;
    const int hf = lane >> 4, l16 = lane & 15;
    int tid = blockIdx.x * 8 + (threadIdx.x >> 5);
    const int tn = T_ / 16, tm = DH1 / 16;
    int b  = tid / (tm * tn);
    int r  = tid % (tm * tn);
    int m0 = (r / tn) * 16;   // c
    int n0 = (r % tn) * 16;   // t

    const float* Arow = w1w + (size_t)(m0 + l16) * DIN;              // A[m][k], row-major
    const float* Brow = x + ((size_t)b * T_ + (n0 + l16)) * DIN;     // B[k][n] = x[b][n][k]
    v8f acc = {};
    #pragma unroll 4
    for (int k0 = 0; k0 < DIN; k0 += 4) {
        v2f a  = *(const v2f*)(Arow + k0 + 2 * hf);
        v2f bv = *(const v2f*)(Brow + k0 + 2 * hf);
        acc = __builtin_amdgcn_wmma_f32_16x16x4_f32(false, a, false, bv,
                                                    (short)0, acc, false, false);
    }
    float* H = h + (size_t)b * DH1 * T_;
    #pragma unroll
    for (int v = 0; v < 8; ++v) {
        int c = m0 + v + 8 * hf;
        int t = n0 + l16;
        H[(size_t)c * T_ + t] = mishf(acc[v] + w1b[c]);
    }
}

// ---------------- depthwise conv + fold gains: y~ = p_g*(conv(h)*d_g + d_b) ----------------
__global__ __launch_bounds__(256) void k_dwconv(
    const float* __restrict__ h, const float* __restrict__ dw,
    const float* __restrict__ dg, const float* __restrict__ db,
    const float* __restrict__ pg, const float* __restrict__ inv_nd,
    float* __restrict__ y)
{
    int idx = blockIdx.x * 256 + threadIdx.x;
    int t = idx & (T_ - 1);
    int c = (idx >> 10) & (DH1 - 1);
    int b = idx >> 20;
    const float* hp  = h + ((size_t)b * DH1 + c) * T_;
    const float* wp  = dw + ((size_t)b * DH1 + c) * KW;
    const float* ndp = inv_nd + b * KW;
    float acc = 0.f;
    #pragma unroll
    for (int k = 0; k < KW; ++k) {
        int tt = t + k - 4;
        if (tt >= 0 && tt < T_)
            acc += hp[tt] * (wp[k] * ndp[k]);
    }
    int bc = b * DH1 + c;
    float yy = acc * dg[bc] + db[bc];
    y[((size_t)b * DH1 + c) * T_ + t] = yy * pg[bc];
}

// ---------------- GEMM 2: z[b][o][t] = (sum_c p_w[b][c][o]*y~[b][c][t]) * inv_np + p_b ----------------
__global__ __launch_bounds__(256) void k_gemm2(
    const float* __restrict__ y, const float* __restrict__ pw,
    const float* __restrict__ pb, const float* __restrict__ inv_np,
    float* __restrict__ z)
{
    const int lane = threadIdx.x & 31;
    const int hf = lane >> 4, l16 = lane & 15;
    int tid = blockIdx.x * 8 + (threadIdx.x >> 5);
    const int tn = T_ / 16, tm = DH2 / 16;
    int b  = tid / (tm * tn);
    int r  = tid % (tm * tn);
    int m0 = (r / tn) * 16;   // o
    int n0 = (r % tn) * 16;   // t
    const float* A  = pw + (size_t)b * DH1 * DH2;  // A[m=o][k=c] = pw[c][o]
    const float* Bm = y  + (size_t)b * DH1 * T_;   // B[k=c][n=t]
    v8f acc = {};
    #pragma unroll 4
    for (int k0 = 0; k0 < DH1; k0 += 4) {
        int ka = k0 + 2 * hf;
        v2f a, bv;
        a.x  = A[(size_t)ka * DH2 + m0 + l16];
        a.y  = A[(size_t)(ka + 1) * DH2 + m0 + l16];
        bv.x = Bm[(size_t)ka * T_ + n0 + l16];
        bv.y = Bm[(size_t)(ka + 1) * T_ + n0 + l16];
        acc = __builtin_amdgcn_wmma_f32_16x16x4_f32(false, a, false, bv,
                                                    (short)0, acc, false, false);
    }
    float* Z = z + (size_t)b * DH2 * T_;
    #pragma unroll
    for (int v = 0; v < 8; ++v) {
        int o = m0 + v + 8 * hf;
        int t = n0 + l16;
        Z[(size_t)o * T_ + t] = acc[v] * inv_np[b * DH2 + o] + pb[b * DH2 + o];
    }
}

// ---------------- GEMM 3: out[b][t][i] = sum_o w2[i][o]*z[b][o][t] + b2[i] + x ----------------
__global__ __launch_bounds__(256) void k_gemm3(
    const float* __restrict__ z, const float* __restrict__ w2w,
    const float* __restrict__ w2b, const float* __restrict__ x,
    float* __restrict__ out)
{
    const int lane = threadIdx.x & 31;
    const int hf = lane >> 4, l16 = lane & 15;
    int tid = blockIdx.x * 8 + (threadIdx.x >> 5);
    const int tn = T_ / 16, tm = DIN / 16;
    int b  = tid / (tm * tn);
    int r  = tid % (tm * tn);
    int m0 = (r / tn) * 16;   // i
    int n0 = (r % tn) * 16;   // t
    const float* Arow = w2w + (size_t)(m0 + l16) * DH2;   // A[m=i][k=o]
    const float* Bm   = z + (size_t)b * DH2 * T_;         // B[k=o][n=t]
    v8f acc = {};
    #pragma unroll 4
    for (int k0 = 0; k0 < DH2; k0 += 4) {
        int ka = k0 + 2 * hf;
        v2f a = *(const v2f*)(Arow + ka);
        v2f bv;
        bv.x = Bm[(size_t)ka * T_ + n0 + l16];
        bv.y = Bm[(size_t)(ka + 1) * T_ + n0 + l16];
        acc = __builtin_amdgcn_wmma_f32_16x16x4_f32(false, a, false, bv,
                                                    (short)0, acc, false, false);
    }
    #pragma unroll
    for (int v = 0; v < 8; ++v) {
        int i = m0 + v + 8 * hf;
        int t = n0 + l16;
        size_t oi = ((size_t)b * T_ + t) * DIN + i;
        out[oi] = acc[v] + w2b[i] + x[oi];
    }
}

extern "C" void kernel_launch(void* const* d_in, const int* in_sizes, int n_in,
                              void* d_out, int out_size, void* d_ws, size_t ws_size,
                              hipStream_t stream)
{
    const float* x   = (const float*)d_in[0];
    const float* d_w = (const float*)d_in[1];
    const float* d_g = (const float*)d_in[2];
    const float* d_b = (const float*)d_in[3];
    const float* p_w = (const float*)d_in[4];
    const float* p_g = (const float*)d_in[5];
    const float* p_b = (const float*)d_in[6];
    const float* w1w = (const float*)d_in[7];
    const float* w1b = (const float*)d_in[8];
    const float* w2w = (const float*)d_in[9];
    const float* w2b = (const float*)d_in[10];
    float* out = (float*)d_out;

    float* ws = (float*)d_ws;
    const size_t NCT = (size_t)B_ * DH1 * T_;   // 8M floats
    float* hbuf   = ws;               // h, then reused for z
    float* ybuf   = ws + NCT;         // y~
    float* inv_nd = ws + 2 * NCT;     // 72 floats (pad to 128)
    float* inv_np = inv_nd + 128;     // 8192 floats

    k_norm_d<<<1, 128, 0, stream>>>(d_w, inv_nd);
    k_norm_p<<<B_ * (DH2 / 64), 256, 0, stream>>>(p_w, inv_np);
    k_gemm1_mish<<<(B_ * (DH1 / 16) * (T_ / 16)) / 8, 256, 0, stream>>>(x, w1w, w1b, hbuf);
    k_dwconv<<<(int)(NCT / 256), 256, 0, stream>>>(hbuf, d_w, d_g, d_b, p_g, inv_nd, ybuf);
    k_gemm2<<<(B_ * (DH2 / 16) * (T_ / 16)) / 8, 256, 0, stream>>>(ybuf, p_w, p_b, inv_np, hbuf);
    k_gemm3<<<(B_ * (DIN / 16) * (T_ / 16)) / 8, 256, 0, stream>>>(hbuf, w2w, w2b, x, out);
}